// RelativeSelfAttention_52682068853379
// MI455X (gfx1250) — compile-verified
//
#include <hip/hip_runtime.h>
#include <hip/hip_bf16.h>
#include <math.h>

// ---------------------------------------------------------------------------
// RelativeSelfAttention forward for MI455X (gfx1250), wave32.
//   B=2, L=2048, D=1024, H=16, hd=64, MAX_REL=128
// f32 WMMA (v_wmma_f32_16x16x4_f32) everywhere; all LDS tile staging uses
// gfx1250 async global->LDS DMA (ASYNCcnt) with double buffering.
// ---------------------------------------------------------------------------

typedef float v2f  __attribute__((ext_vector_type(2)));
typedef float v8f  __attribute__((ext_vector_type(8)));

#define B_   2
#define L_   2048
#define D_   1024
#define H_   16
#define HD_  64
#define MAXREL_ 128
#define NEG_INF_ (-1e30f)
#define SCALE_   (0.125f)           // 1/sqrt(64)

__device__ __forceinline__ v8f wmma4(v2f a, v2f b, v8f c) {
  return __builtin_amdgcn_wmma_f32_16x16x4_f32(
      false, a, false, b, (short)0, c, false, false);
}

// --- gfx1250 async global->LDS DMA (tracked by ASYNCcnt) -------------------
__device__ __forceinline__ unsigned lds_off(const void* p) {
  // generic LDS pointer: addr[31:0] is the LDS byte address (ISA 10.2)
  return (unsigned)(size_t)p;
}
__device__ __forceinline__ void async_load_b128(unsigned lds, const void* g) {
  asm volatile("global_load_async_to_lds_b128 %0, %1, off"
               :: "v"(lds), "v"(g) : "memory");
}
__device__ __forceinline__ void async_load_b32(unsigned lds, const void* g) {
  asm volatile("global_load_async_to_lds_b32 %0, %1, off"
               :: "v"(lds), "v"(g) : "memory");
}
template <int N>
__device__ __forceinline__ void async_wait_le() {
  asm volatile("s_wait_asynccnt %0" :: "i"(N) : "memory");
}

// ===========================================================================
// Kernel 1: QKV projection. C[4096,3072] = X[4096,1024] @ Wqkv + bqkv
// 64x64 tile/WG, 8 waves, wave -> 16 rows x 32 cols (two 16x16 accumulators).
// A tile async b128; B tile staged TRANSPOSED (Bt[n][k]) via async b32 so
// every WMMA fragment load is a contiguous 8-byte LDS read. Double-buffered.
// ===========================================================================
__global__ __launch_bounds__(256) void qkv_gemm_kernel(
    const float* __restrict__ X, const float* __restrict__ W,
    const float* __restrict__ bias,
    float* __restrict__ q_ws, float* __restrict__ k_ws, float* __restrict__ v_ws)
{
  const int m0 = blockIdx.y * 64;
  const int n0 = blockIdx.x * 64;
  const int tid  = threadIdx.x;
  const int lane = tid & 31;
  const int wave = tid >> 5;
  const int rw = wave >> 1;
  const int cw = wave & 1;
  const int lhi = lane >> 4;
  const int mA = lane & 15;
  const int kA = lhi << 1;
  const int nB = lane & 15;
  const int kB = lhi << 1;

  __shared__ float As[2][64][36];      // A[m][k]
  __shared__ float Bt[2][64][36];      // B transposed: [n][k]

  v8f acc0 = {}, acc1 = {};

  auto stage = [&](int buf, int kk) {
    #pragma unroll
    for (int p = 0; p < 2; ++p) {      // A: 512 x b128
      int idx = tid + p * 256;
      int row = idx >> 3;
      int c4  = (idx & 7) << 2;
      async_load_b128(lds_off(&As[buf][row][c4]),
                      &X[(size_t)(m0 + row) * D_ + kk + c4]);
    }
    #pragma unroll
    for (int p = 0; p < 8; ++p) {      // B: 2048 x b32, transposed into LDS
      int idx = tid + p * 256;
      int k   = idx >> 6;              // 0..31
      int n   = idx & 63;              // coalesced along n
      async_load_b32(lds_off(&Bt[buf][n][k]),
                     &W[(size_t)(kk + k) * (3 * D_) + n0 + n]);
    }
  };

  stage(0, 0);
  for (int step = 0; step < D_ / 32; ++step) {
    int buf = step & 1;
    if (step + 1 < D_ / 32) { stage(buf ^ 1, (step + 1) * 32); async_wait_le<10>(); }
    else                    { async_wait_le<0>(); }
    __syncthreads();
    #pragma unroll
    for (int ks = 0; ks < 32; ks += 4) {
      v2f a  = *(const v2f*)&As[buf][rw * 16 + mA][ks + kA];
      v2f b0 = *(const v2f*)&Bt[buf][cw * 32 + nB][ks + kB];
      v2f b1 = *(const v2f*)&Bt[buf][cw * 32 + 16 + nB][ks + kB];
      acc0 = wmma4(a, b0, acc0);
      acc1 = wmma4(a, b1, acc1);
    }
    __syncthreads();
  }

  #pragma unroll
  for (int f = 0; f < 2; ++f) {
    v8f acc = f ? acc1 : acc0;
    int n = n0 + cw * 32 + f * 16 + nB;
    int which = n >> 10;
    int d     = n & 1023;
    int h     = d >> 6;
    int dd    = d & 63;
    float bv  = bias[n];
    float* dst = (which == 0) ? q_ws : ((which == 1) ? k_ws : v_ws);
    #pragma unroll
    for (int r = 0; r < 8; ++r) {
      int m = m0 + rw * 16 + r + 8 * lhi;
      int b = m >> 11;
      int l = m & 2047;
      dst[(((size_t)b * H_ + h) * L_ + l) * HD_ + dd] = acc[r] + bv;
    }
  }
}

// ===========================================================================
// Kernel 2: flash attention. One WG per (64-row q block, head, batch).
// K/V blocks double-buffered through async DMA; online softmax in LDS.
// ===========================================================================
__global__ __launch_bounds__(256) void flash_attn_kernel(
    const float* __restrict__ q_ws, const float* __restrict__ k_ws,
    const float* __restrict__ v_ws, const unsigned char* __restrict__ mask,
    const float* __restrict__ bias_table, float* __restrict__ attn_ws)
{
  const int qb = blockIdx.x;
  const int h  = blockIdx.y;
  const int b  = blockIdx.z;
  const int tid  = threadIdx.x;
  const int lane = tid & 31;
  const int wave = tid >> 5;
  const int rw = wave >> 1;
  const int cw = wave & 1;
  const int lhi = lane >> 4;
  const int mA = lane & 15;
  const int kA = lhi << 1;
  const int nB = lane & 15;
  const int kB = lhi << 1;
  const int NB = L_ / 64;

  __shared__ float Qs[64][68];
  __shared__ float Ks[2][64][68];
  __shared__ float Vs[2][64][68];
  __shared__ float Ss[64][68];
  __shared__ float row_m[64], row_l[64], row_alpha[64];

  const size_t head_base = ((size_t)b * H_ + h) * L_;

  // Q block: 4 async b128 per thread
  #pragma unroll
  for (int p = 0; p < 4; ++p) {
    int idx = tid + p * 256;
    int row = idx >> 4;
    int c4  = (idx & 15) << 2;
    async_load_b128(lds_off(&Qs[row][c4]),
                    &q_ws[(head_base + qb * 64 + row) * HD_ + c4]);
  }

  auto stage_kv = [&](int buf, int kb) {
    #pragma unroll
    for (int p = 0; p < 4; ++p) {
      int idx = tid + p * 256;
      int row = idx >> 4;
      int c4  = (idx & 15) << 2;
      size_t src = (head_base + (size_t)kb * 64 + row) * HD_ + c4;
      async_load_b128(lds_off(&Ks[buf][row][c4]), &k_ws[src]);
      async_load_b128(lds_off(&Vs[buf][row][c4]), &v_ws[src]);
    }
  };

  if (tid < 64) { row_m[tid] = -3.0e38f; row_l[tid] = 0.0f; }

  v8f o0 = {}, o1 = {};
  stage_kv(0, 0);

  for (int kb = 0; kb < NB; ++kb) {
    int buf = kb & 1;
    if (kb + 1 < NB) { stage_kv(buf ^ 1, kb + 1); async_wait_le<8>(); }
    else             { async_wait_le<0>(); }
    __syncthreads();

    // ---- S = Q Kt ----
    v8f s0 = {}, s1 = {};
    #pragma unroll
    for (int ks = 0; ks < HD_; ks += 4) {
      v2f a  = *(const v2f*)&Qs[rw * 16 + mA][ks + kA];
      v2f b0 = *(const v2f*)&Ks[buf][cw * 32 + nB][ks + kB];       // B[k][n]=K[n][k]
      v2f b1 = *(const v2f*)&Ks[buf][cw * 32 + 16 + nB][ks + kB];
      s0 = wmma4(a, b0, s0);
      s1 = wmma4(a, b1, s1);
    }
    #pragma unroll
    for (int r = 0; r < 8; ++r) {
      int mrow = rw * 16 + r + 8 * lhi;
      Ss[mrow][cw * 32 + nB]      = s0[r] * SCALE_;
      Ss[mrow][cw * 32 + 16 + nB] = s1[r] * SCALE_;
    }
    __syncthreads();

    // ---- per-row rel-bias + mask + online softmax stats ----
    if (tid < 64) {
      int qg = qb * 64 + tid;
      float mold = row_m[tid];
      float mmax = mold;
      for (int j = 0; j < 64; ++j) {
        int kg  = kb * 64 + j;
        int rel = kg - qg;
        rel = rel < -MAXREL_ ? -MAXREL_ : (rel > MAXREL_ ? MAXREL_ : rel);
        float s = Ss[tid][j] + bias_table[(rel + MAXREL_) * H_ + h];
        if (mask[(size_t)b * L_ + kg]) s = NEG_INF_;
        Ss[tid][j] = s;
        mmax = fmaxf(mmax, s);
      }
      float alpha = __expf(mold - mmax);
      float lsum = 0.0f;
      for (int j = 0; j < 64; ++j) {
        float p = __expf(Ss[tid][j] - mmax);
        Ss[tid][j] = p;
        lsum += p;
      }
      row_l[tid] = alpha * row_l[tid] + lsum;
      row_m[tid] = mmax;
      row_alpha[tid] = alpha;
    }
    __syncthreads();

    // ---- O = alpha*O + P V ----
    #pragma unroll
    for (int r = 0; r < 8; ++r) {
      float al = row_alpha[rw * 16 + r + 8 * lhi];
      o0[r] *= al;
      o1[r] *= al;
    }
    #pragma unroll
    for (int ns = 0; ns < 64; ns += 4) {
      v2f a = *(const v2f*)&Ss[rw * 16 + mA][ns + kA];
      v2f b0, b1;
      b0.x = Vs[buf][ns + kB][cw * 32 + nB];
      b0.y = Vs[buf][ns + kB + 1][cw * 32 + nB];
      b1.x = Vs[buf][ns + kB][cw * 32 + 16 + nB];
      b1.y = Vs[buf][ns + kB + 1][cw * 32 + 16 + nB];
      o0 = wmma4(a, b0, o0);
      o1 = wmma4(a, b1, o1);
    }
    __syncthreads();
  }

  #pragma unroll
  for (int r = 0; r < 8; ++r) {
    int mrow = rw * 16 + r + 8 * lhi;
    int l    = qb * 64 + mrow;
    float inv = 1.0f / row_l[mrow];
    int d0 = cw * 32 + nB;
    attn_ws[((size_t)b * L_ + l) * D_ + h * HD_ + d0]      = o0[r] * inv;
    attn_ws[((size_t)b * L_ + l) * D_ + h * HD_ + d0 + 16] = o1[r] * inv;
  }
}

// ===========================================================================
// Kernel 3: output projection. Y[4096,1024] = A @ Wout + bout
// ===========================================================================
__global__ __launch_bounds__(256) void out_gemm_kernel(
    const float* __restrict__ A, const float* __restrict__ W,
    const float* __restrict__ bias, float* __restrict__ Y)
{
  const int m0 = blockIdx.y * 64;
  const int n0 = blockIdx.x * 64;
  const int tid  = threadIdx.x;
  const int lane = tid & 31;
  const int wave = tid >> 5;
  const int rw = wave >> 1;
  const int cw = wave & 1;
  const int lhi = lane >> 4;
  const int mA = lane & 15;
  const int kA = lhi << 1;
  const int nB = lane & 15;
  const int kB = lhi << 1;

  __shared__ float As[2][64][36];
  __shared__ float Bt[2][64][36];

  v8f acc0 = {}, acc1 = {};

  auto stage = [&](int buf, int kk) {
    #pragma unroll
    for (int p = 0; p < 2; ++p) {
      int idx = tid + p * 256;
      int row = idx >> 3;
      int c4  = (idx & 7) << 2;
      async_load_b128(lds_off(&As[buf][row][c4]),
                      &A[(size_t)(m0 + row) * D_ + kk + c4]);
    }
    #pragma unroll
    for (int p = 0; p < 8; ++p) {
      int idx = tid + p * 256;
      int k   = idx >> 6;
      int n   = idx & 63;
      async_load_b32(lds_off(&Bt[buf][n][k]),
                     &W[(size_t)(kk + k) * D_ + n0 + n]);
    }
  };

  stage(0, 0);
  for (int step = 0; step < D_ / 32; ++step) {
    int buf = step & 1;
    if (step + 1 < D_ / 32) { stage(buf ^ 1, (step + 1) * 32); async_wait_le<10>(); }
    else                    { async_wait_le<0>(); }
    __syncthreads();
    #pragma unroll
    for (int ks = 0; ks < 32; ks += 4) {
      v2f a  = *(const v2f*)&As[buf][rw * 16 + mA][ks + kA];
      v2f b0 = *(const v2f*)&Bt[buf][cw * 32 + nB][ks + kB];
      v2f b1 = *(const v2f*)&Bt[buf][cw * 32 + 16 + nB][ks + kB];
      acc0 = wmma4(a, b0, acc0);
      acc1 = wmma4(a, b1, acc1);
    }
    __syncthreads();
  }

  #pragma unroll
  for (int f = 0; f < 2; ++f) {
    v8f acc = f ? acc1 : acc0;
    int n = n0 + cw * 32 + f * 16 + nB;
    float bv = bias[n];
    #pragma unroll
    for (int r = 0; r < 8; ++r) {
      int m = m0 + rw * 16 + r + 8 * lhi;
      Y[(size_t)m * D_ + n] = acc[r] + bv;
    }
  }
}

// ===========================================================================
// Launch
// ===========================================================================
extern "C" void kernel_launch(void* const* d_in, const int* in_sizes, int n_in,
                              void* d_out, int out_size, void* d_ws, size_t ws_size,
                              hipStream_t stream) {
  const float*         x     = (const float*)d_in[0];
  const unsigned char* kpm   = (const unsigned char*)d_in[1];  // bool [B,L]
  const float*         Wqkv  = (const float*)d_in[2];
  const float*         bqkv  = (const float*)d_in[3];
  const float*         Wout  = (const float*)d_in[4];
  const float*         bout  = (const float*)d_in[5];
  const float*         btab  = (const float*)d_in[6];
  float*               out   = (float*)d_out;

  const size_t per = (size_t)B_ * H_ * L_ * HD_;   // 4,194,304 floats
  float* ws   = (float*)d_ws;
  float* q_ws = ws;
  float* k_ws = ws + per;
  float* v_ws = ws + 2 * per;
  float* a_ws = ws + 3 * per;                      // [B, L, D]

  qkv_gemm_kernel<<<dim3(3 * D_ / 64, B_ * L_ / 64), 256, 0, stream>>>(
      x, Wqkv, bqkv, q_ws, k_ws, v_ws);

  flash_attn_kernel<<<dim3(L_ / 64, H_, B_), 256, 0, stream>>>(
      q_ws, k_ws, v_ws, kpm, btab, a_ws);

  out_gemm_kernel<<<dim3(D_ / 64, B_ * L_ / 64), 256, 0, stream>>>(
      a_ws, Wout, bout, out);
}